// GATLayer_48000554500645
// MI455X (gfx1250) — compile-verified
//
#include <hip/hip_runtime.h>
#include <cstdint>

#define N_NODES 50000
#define N_EDGES 600000
#define DIM 128

typedef __attribute__((ext_vector_type(2))) float v2f;
typedef __attribute__((ext_vector_type(8))) float v8f;

// Monotone order-preserving float<->int mapping for atomicMax-based f32 max.
__device__ __forceinline__ int f32_to_ordered(float f) {
  int i = __float_as_int(f);
  return (i >= 0) ? i : (i ^ 0x7FFFFFFF);
}
__device__ __forceinline__ float ordered_to_f32(int i) {
  int b = (i >= 0) ? i : (i ^ 0x7FFFFFFF);
  return __int_as_float(b);
}

// K0: zero output accumulator, init segment max (-inf ordering) and segment sum.
__global__ void k0_init(float* __restrict__ out, int* __restrict__ seg_max_i,
                        float* __restrict__ seg_sum) {
  const int t = blockIdx.x * blockDim.x + threadIdx.x;   // grid == N_NODES*DIM exactly
  out[t] = 0.0f;
  if (t < N_NODES) {
    seg_max_i[t] = (int)0x80000000;  // ordered -inf
    seg_sum[t]   = 0.0f;
  }
}

// K1: one wave = 16 edges. WMMA f32 16x16x4 computes the three 128-length dots:
//   accP col0 = (ft_s*ft_d*h_d) . W_pi                       (chain P)
//   accQ col0 = (ft_s*ft_d).W_M[0:128] + h_d.W_M[128:256]    (chain Q, shared acc)
// A layout (16x4 f32): lane L(<16) = edge L, K={0,1}; lane L+16 = edge L, K={2,3}.
// B layout (4x16 f32): col 0 lives in lanes 0 (K=0,1) and 16 (K=2,3); other
// columns must be zero -> lanes with r!=0 read from a zeroed LDS pad, so the
// inner loop needs NO predication (unconditional ds_load_b64, full EXEC).
__global__ void __launch_bounds__(256) k1_logits(
    const float* __restrict__ h_v, const float* __restrict__ h_d,
    const float* __restrict__ W_pi, const float* __restrict__ W_M,
    const int* __restrict__ src, const int* __restrict__ dst,
    float* __restrict__ logits, int* __restrict__ seg_max_i) {
  // [0,128)=W_pi  [128,384)=W_M  [384,512)=zero pad for non-column-0 lanes
  __shared__ __align__(16) float sW[4 * DIM];
  for (int i = threadIdx.x; i < 4 * DIM; i += 256) {
    float v = 0.0f;
    if (i < DIM)          v = W_pi[i];
    else if (i < 3 * DIM) v = W_M[i - DIM];
    sW[i] = v;
  }
  __syncthreads();

  const int wid  = threadIdx.x >> 5;
  const int lane = threadIdx.x & 31;
  const int r    = lane & 15;   // edge row in tile
  const int half = lane >> 4;   // K-pair select
  const long long edgeBase = ((long long)blockIdx.x * 8 + wid) * 16;
  if (edgeBase >= N_EDGES) return;             // wave-uniform: EXEC stays all-1s

  const int e = (int)edgeBase + r;
  const int s = src[e];
  const int d = dst[e];
  const float* ps = h_v + (size_t)s * DIM + half * 2;
  const float* pd = h_v + (size_t)d * DIM + half * 2;
  const float* ph = h_d + (size_t)e * DIM + half * 2;

  // Hoisted pointer select: lanes 0/16 read real weights, all others read zeros.
  const bool bsel = (r == 0);
  const float* wp  = sW + (bsel ? 0       : 3 * DIM) + half * 2;
  const float* wml = sW + (bsel ? DIM     : 3 * DIM) + half * 2;
  const float* wmh = sW + (bsel ? 2 * DIM : 3 * DIM) + half * 2;

  v8f accP = {};
  v8f accQ = {};
#pragma unroll 4
  for (int kk = 0; kk < 32; ++kk) {
    const int off = kk * 4;
    v2f fs = *(const v2f*)(ps + off);
    v2f fd = *(const v2f*)(pd + off);
    v2f hd = *(const v2f*)(ph + off);
    v2f bp  = *(const v2f*)(wp + off);
    v2f bq1 = *(const v2f*)(wml + off);
    v2f bq2 = *(const v2f*)(wmh + off);
    v2f prod = fs * fd;
    v2f php  = prod * hd;
    accP = __builtin_amdgcn_wmma_f32_16x16x4_f32(false, php,  false, bp,  (short)0, accP, false, false);
    accQ = __builtin_amdgcn_wmma_f32_16x16x4_f32(false, prod, false, bq1, (short)0, accQ, false, false);
    accQ = __builtin_amdgcn_wmma_f32_16x16x4_f32(false, hd,   false, bq2, (short)0, accQ, false, false);
  }

  // D col0: lane 0 holds edges 0..7 in acc[0..7]; lane 16 holds edges 8..15.
  if (r == 0) {
    const int rbase = (int)edgeBase + half * 8;
#pragma unroll
    for (int j = 0; j < 8; ++j) {
      const float d1 = accP[j];
      const float q  = accQ[j];
      const float m  = 1.0f / (1.0f + __expf(-q));   // sigmoid gate
      const float lg = d1 * m;
      const int e2 = rbase + j;
      logits[e2] = lg;
      atomicMax(seg_max_i + dst[e2], f32_to_ordered(lg));
    }
  }
}

// K2: one wave per edge. ex = exp(logit - segmax[dst]); accumulate seg_sum and
// the un-normalized message sum out[dst] += ft_src * ex (L2-resident atomics).
__global__ void __launch_bounds__(256) k2_accum(
    const float* __restrict__ h_v, const int* __restrict__ src,
    const int* __restrict__ dst, const float* __restrict__ logits,
    const int* __restrict__ seg_max_i, float* __restrict__ seg_sum,
    float* __restrict__ out) {
  const int wid  = threadIdx.x >> 5;
  const int lane = threadIdx.x & 31;
  const int e = blockIdx.x * 8 + wid;          // grid == N_EDGES/8 exactly
  const int dd = dst[e];
  const float lg = logits[e];
  const float mx = ordered_to_f32(seg_max_i[dd]);
  const float ex = __expf(lg - mx);
  if (lane == 0) atomicAdd(seg_sum + dd, ex);
  const int s = src[e];
  const float4 fs = *((const float4*)(h_v + (size_t)s * DIM) + lane);
  float* op = out + (size_t)dd * DIM + lane * 4;
  atomicAdd(op + 0, fs.x * ex);
  atomicAdd(op + 1, fs.y * ex);
  atomicAdd(op + 2, fs.z * ex);
  atomicAdd(op + 3, fs.w * ex);
}

// K3: per-node normalization; empty segments (seg_sum==0) output zero.
__global__ void k3_norm(float* __restrict__ out, const float* __restrict__ seg_sum) {
  const int t = blockIdx.x * blockDim.x + threadIdx.x;   // grid == N_NODES*32 exactly
  const int v = t >> 5;
  const int q = t & 31;
  const float ss  = seg_sum[v];
  const float inv = (ss > 0.0f) ? (1.0f / ss) : 0.0f;
  float4* po = (float4*)(out + (size_t)v * DIM) + q;
  float4 o = *po;
  o.x *= inv; o.y *= inv; o.z *= inv; o.w *= inv;
  *po = o;
}

extern "C" void kernel_launch(void* const* d_in, const int* in_sizes, int n_in,
                              void* d_out, int out_size, void* d_ws, size_t ws_size,
                              hipStream_t stream) {
  const float* h_v  = (const float*)d_in[0];
  const float* h_d  = (const float*)d_in[1];
  const float* W_pi = (const float*)d_in[2];
  const float* W_M  = (const float*)d_in[3];
  const int*   src  = (const int*)d_in[4];
  const int*   dst  = (const int*)d_in[5];
  float* out = (float*)d_out;

  float* logits    = (float*)d_ws;                     // 600000 f32
  int*   seg_max_i = (int*)(logits + N_EDGES);         // 50000 i32
  float* seg_sum   = (float*)(seg_max_i + N_NODES);    // 50000 f32  (~2.8 MB total)

  k0_init<<<(N_NODES * DIM) / 256, 256, 0, stream>>>(out, seg_max_i, seg_sum);
  k1_logits<<<((N_EDGES / 16) + 7) / 8, 256, 0, stream>>>(h_v, h_d, W_pi, W_M, src, dst,
                                                          logits, seg_max_i);
  k2_accum<<<N_EDGES / 8, 256, 0, stream>>>(h_v, src, dst, logits, seg_max_i, seg_sum, out);
  k3_norm<<<(N_NODES * 32) / 256, 256, 0, stream>>>(out, seg_sum);
}